// DLMPNN_45148696215938
// MI455X (gfx1250) — compile-verified
//
#include <hip/hip_runtime.h>
#include <math.h>

#define EDIM 128

typedef float v2f __attribute__((ext_vector_type(2)));
typedef float v8f __attribute__((ext_vector_type(8)));

static __device__ __forceinline__ float silu_f(float v) {
  return v / (1.0f + __expf(-v));
}

// ---------------------------------------------------------------------------
// Pack W[128][128] (row-major k,n) into WMMA B-fragment pair order:
//   WP2[kp*128 + n] = { W[2*kp][n], W[2*kp+1][n] },  kp = 0..63
// so a B fragment (k0 = kk+2*khalf) is one float2 load at (k0/2)*128 + n.
// ---------------------------------------------------------------------------
__global__ void pack_w_kernel(const float* __restrict__ W, float* __restrict__ WP)
{
  const int i = blockIdx.x * blockDim.x + threadIdx.x;   // 0..8191
  if (i >= 64 * EDIM) return;
  const int kp = i >> 7;
  const int n  = i & 127;
  float2 p;
  p.x = W[(size_t)(2 * kp + 0) * EDIM + n];
  p.y = W[(size_t)(2 * kp + 1) * EDIM + n];
  ((float2*)WP)[i] = p;
}

// ---------------------------------------------------------------------------
// WMMA fp32 GEMM: S = A[gather(row)] @ W  (A rows [*,128], WP packed pairs)
// epilogue modes:
//   0: C = S
//   1: C = C + S
//   2: C = silu(S + bias)
//   3: C = aux + silu(S + bias)
//   4: C = silu(C + S + bias)
// 256 threads = 8 waves; block covers 64 rows x 128 cols.
// wave w: row tile 16*(w&3), col half 64*(w>>2), 4 col tiles of 16.
// ---------------------------------------------------------------------------
__global__ __launch_bounds__(256) void gemm128_wmma_kernel(
    const float* __restrict__ A, const int* __restrict__ aidx,
    const float* __restrict__ WPf, const float* __restrict__ bias,
    float* __restrict__ C, const float* __restrict__ aux,
    int nrows, int mode)
{
  __shared__ float sA[64][EDIM + 4];

  const int tid = threadIdx.x;
  const int rowBase = blockIdx.x * 64;

  // stage 64 (possibly gathered) A rows into LDS, float4 at a time
  for (int i = tid * 4; i < 64 * EDIM; i += 256 * 4) {
    const int r = i >> 7;
    const int c = i & 127;
    float4 a4 = make_float4(0.f, 0.f, 0.f, 0.f);
    const int gr = rowBase + r;
    if (gr < nrows) {
      const int src = aidx ? aidx[gr] : gr;
      a4 = *(const float4*)(A + (size_t)src * EDIM + c);
    }
    sA[r][c + 0] = a4.x; sA[r][c + 1] = a4.y;
    sA[r][c + 2] = a4.z; sA[r][c + 3] = a4.w;
  }
  __syncthreads();

  const int wave     = tid >> 5;
  const int lane     = tid & 31;
  const int mrow     = 16 * (wave & 3);
  const int ncolBase = 64 * (wave >> 2);
  const int m        = lane & 15;   // M for A frag / N for B frag / N for C
  const int khalf    = lane >> 4;   // K-half selector (A/B), M-offset (C)

  const float2* WP2 = (const float2*)WPf;

  v8f acc[4];
  acc[0] = v8f{}; acc[1] = v8f{}; acc[2] = v8f{}; acc[3] = v8f{};

  for (int kk = 0; kk < EDIM; kk += 4) {
    const int k0 = kk + 2 * khalf;
    v2f a;
    a.x = sA[mrow + m][k0];
    a.y = sA[mrow + m][k0 + 1];
    const float2* wp = WP2 + (size_t)(k0 >> 1) * EDIM + ncolBase + m;
#pragma unroll
    for (int t = 0; t < 4; ++t) {
      const float2 bp = wp[16 * t];
      v2f b;
      b.x = bp.x;
      b.y = bp.y;
      acc[t] = __builtin_amdgcn_wmma_f32_16x16x4_f32(
          false, a, false, b, (short)0, acc[t], false, false);
    }
  }

  // epilogue: C/D layout: VGPR i, lanes 0-15 -> M=i, lanes 16-31 -> M=i+8; N=lane&15
#pragma unroll
  for (int t = 0; t < 4; ++t) {
    const int n = ncolBase + 16 * t + m;
    const float bv = bias ? bias[n] : 0.0f;
#pragma unroll
    for (int i = 0; i < 8; ++i) {
      const int r = rowBase + mrow + i + 8 * khalf;
      if (r < nrows) {
        const float s = acc[t][i];
        float* cp = C + (size_t)r * EDIM + n;
        if (mode == 0)      *cp = s;
        else if (mode == 1) *cp = *cp + s;
        else if (mode == 2) *cp = silu_f(s + bv);
        else if (mode == 3) *cp = aux[(size_t)r * EDIM + n] + silu_f(s + bv);
        else                *cp = silu_f(*cp + s + bv);
      }
    }
  }
}

// ---------------------------------------------------------------------------
// per-edge Bessel radial basis: rbf[e][k] = sqrt(2/5)*sin((k+1)*pi*d/5)/d
// ---------------------------------------------------------------------------
__global__ void edge_rbf_kernel(const float* __restrict__ R,
                                const int* __restrict__ idx_i,
                                const int* __restrict__ idx_j,
                                float* __restrict__ rbf, int ne)
{
  const int e = blockIdx.x * blockDim.x + threadIdx.x;
  if (e >= ne) return;
  const int i = idx_i[e], j = idx_j[e];
  const float dx = R[3 * i + 0] - R[3 * j + 0];
  const float dy = R[3 * i + 1] - R[3 * j + 1];
  const float dz = R[3 * i + 2] - R[3 * j + 2];
  float d = sqrtf(dx * dx + dy * dy + dz * dz);
  d = fmaxf(d, 1e-2f);
  const float s  = 0.6324555320336759f / d;            // sqrt(2/cutoff)/d
  const float ph = d * (3.14159265358979323846f / 5.0f);
#pragma unroll
  for (int k = 0; k < 6; ++k)
    rbf[(size_t)e * 6 + k] = s * sinf((float)(k + 1) * ph);
}

// gather 128-float rows: dst[a] = src[idx[a]]   (32 lanes x float4 per row)
__global__ void gather_rows_kernel(const float* __restrict__ src,
                                   const int* __restrict__ idx,
                                   float* __restrict__ dst, int n)
{
  const int gid = blockIdx.x * blockDim.x + threadIdx.x;
  const int a = gid >> 5, lane = gid & 31;
  if (a >= n) return;
  const int c = lane * 4;
  *(float4*)(dst + (size_t)a * EDIM + c) =
      *(const float4*)(src + (size_t)idx[a] * EDIM + c);
}

// ---------------------------------------------------------------------------
// small-K radial projection: dot[c] = sum_{k<6} rbf[e][k] * Wk[k][c]
//   mode 0: Out[e][c]  = silu(dot)                 (rbf_e)
//   mode 1: Out[e][c]  = Xin[e][c] * dot           (x_kj *= rbf@W_irbf)
//   mode 2: atomicAdd(Out[sidx[e]][c], Xin[e][c]*dot)   (output-block scatter)
// 32 lanes per edge, float4 per lane.
// ---------------------------------------------------------------------------
__global__ __launch_bounds__(256) void edge_radial_kernel(
    const float* __restrict__ rbf, const float* __restrict__ Wk,
    const float* __restrict__ Xin, const int* __restrict__ sidx,
    float* __restrict__ Out, int ne, int mode)
{
  __shared__ float sW[6 * EDIM];
  for (int i = threadIdx.x; i < 6 * EDIM; i += 256) sW[i] = Wk[i];
  __syncthreads();

  const int gid = blockIdx.x * 256 + threadIdx.x;
  const int e = gid >> 5, lane = gid & 31;
  if (e >= ne) return;

  float rb[6];
#pragma unroll
  for (int k = 0; k < 6; ++k) rb[k] = rbf[(size_t)e * 6 + k];

  const int c = lane * 4;
  float d0 = 0.f, d1 = 0.f, d2 = 0.f, d3 = 0.f;
#pragma unroll
  for (int k = 0; k < 6; ++k) {
    const float* wr = sW + k * EDIM + c;
    d0 = fmaf(rb[k], wr[0], d0);
    d1 = fmaf(rb[k], wr[1], d1);
    d2 = fmaf(rb[k], wr[2], d2);
    d3 = fmaf(rb[k], wr[3], d3);
  }

  if (mode == 0) {
    float4 o;
    o.x = silu_f(d0); o.y = silu_f(d1); o.z = silu_f(d2); o.w = silu_f(d3);
    *(float4*)(Out + (size_t)e * EDIM + c) = o;
  } else if (mode == 1) {
    float4 x4 = *(const float4*)(Xin + (size_t)e * EDIM + c);
    float4 o;
    o.x = x4.x * d0; o.y = x4.y * d1; o.z = x4.z * d2; o.w = x4.w * d3;
    *(float4*)(Out + (size_t)e * EDIM + c) = o;
  } else {
    float4 x4 = *(const float4*)(Xin + (size_t)e * EDIM + c);
    float* dst = Out + (size_t)sidx[e] * EDIM + c;
    atomicAdd(dst + 0, x4.x * d0);
    atomicAdd(dst + 1, x4.y * d1);
    atomicAdd(dst + 2, x4.z * d2);
    atomicAdd(dst + 3, x4.w * d3);
  }
}

// ---------------------------------------------------------------------------
// triplet message: m = x_kj[idx_kj] * (abf @ W_abf), aggr[idx_ji] += m
// abf_k = cos(k*theta) = Chebyshev T_k(clip(cos)). 32 lanes per triplet.
// ---------------------------------------------------------------------------
__global__ __launch_bounds__(256) void triplet_kernel(
    const float* __restrict__ XKJ, const float* __restrict__ Wabf, /*[7,128]*/
    const float* __restrict__ cosine,
    const int* __restrict__ idx_kj, const int* __restrict__ idx_ji,
    float* __restrict__ AGGR, int nt)
{
  __shared__ float sW[7 * EDIM];
  for (int i = threadIdx.x; i < 7 * EDIM; i += 256) sW[i] = Wabf[i];
  __syncthreads();

  const int gid = blockIdx.x * 256 + threadIdx.x;
  const int t = gid >> 5, lane = gid & 31;
  if (t >= nt) return;

  float cth = cosine[t];
  cth = fminf(1.0f, fmaxf(-1.0f, cth));
  float tk[7];
  tk[0] = 1.0f;
  tk[1] = cth;
#pragma unroll
  for (int k = 2; k < 7; ++k) tk[k] = 2.0f * cth * tk[k - 1] - tk[k - 2];

  const int kj = idx_kj[t];
  const int ji = idx_ji[t];
  const int c = lane * 4;

  const float4 x4 = *(const float4*)(XKJ + (size_t)kj * EDIM + c);
  float w0 = 0.f, w1 = 0.f, w2 = 0.f, w3 = 0.f;
#pragma unroll
  for (int k = 0; k < 7; ++k) {
    const float* wr = sW + k * EDIM + c;
    w0 = fmaf(tk[k], wr[0], w0);
    w1 = fmaf(tk[k], wr[1], w1);
    w2 = fmaf(tk[k], wr[2], w2);
    w3 = fmaf(tk[k], wr[3], w3);
  }
  float* dst = AGGR + (size_t)ji * EDIM + c;
  atomicAdd(dst + 0, x4.x * w0);
  atomicAdd(dst + 1, x4.y * w1);
  atomicAdd(dst + 2, x4.z * w2);
  atomicAdd(dst + 3, x4.w * w3);
}

// dst[i] += src[i], float4 granularity
__global__ void ew_add_kernel(float* __restrict__ dst,
                              const float* __restrict__ src, int n4)
{
  const int i = blockIdx.x * blockDim.x + threadIdx.x;
  if (i >= n4) return;
  float4 a = ((const float4*)src)[i];
  float4 b = ((float4*)dst)[i];
  b.x += a.x; b.y += a.y; b.z += a.z; b.w += a.w;
  ((float4*)dst)[i] = b;
}

// final molecule reduction: out[seg[a]] += cmp*res_out[a] + csg*res_sing[a]
__global__ void final_kernel(const float* __restrict__ ro,
                             const float* __restrict__ rs,
                             const int* __restrict__ seg,
                             const float* __restrict__ cmp,
                             const float* __restrict__ csg,
                             float* __restrict__ out, int na)
{
  const int gid = blockIdx.x * blockDim.x + threadIdx.x;
  const int a = gid >> 5, lane = gid & 31;
  if (a >= na) return;
  const int c = lane * 4;
  const float km = cmp[0], ks = csg[0];
  const float4 o4 = *(const float4*)(ro + (size_t)a * EDIM + c);
  const float4 s4 = *(const float4*)(rs + (size_t)a * EDIM + c);
  float* dst = out + (size_t)seg[a] * EDIM + c;
  atomicAdd(dst + 0, km * o4.x + ks * s4.x);
  atomicAdd(dst + 1, km * o4.y + ks * s4.y);
  atomicAdd(dst + 2, km * o4.z + ks * s4.z);
  atomicAdd(dst + 3, km * o4.w + ks * s4.w);
}

static inline int cdiv_i(long a, long b) { return (int)((a + b - 1) / b); }

extern "C" void kernel_launch(void* const* d_in, const int* in_sizes, int n_in,
                              void* d_out, int out_size, void* d_ws, size_t ws_size,
                              hipStream_t stream)
{
  (void)n_in; (void)ws_size;
  const int*   Z         = (const int*)  d_in[0];
  const float* R         = (const float*)d_in[1];
  const int*   batch_seg = (const int*)  d_in[2];
  const int*   idx_i     = (const int*)  d_in[3];
  const int*   idx_j     = (const int*)  d_in[4];
  const int*   idx_kj    = (const int*)  d_in[5];
  const int*   idx_ji    = (const int*)  d_in[6];
  const float* cosine    = (const float*)d_in[7];
  const float* emb_atom  = (const float*)d_in[8];
  const float* W_rbf_emb = (const float*)d_in[9];
  const float* W_emb     = (const float*)d_in[10];
  const float* b_emb     = (const float*)d_in[11];
  const float* W_orbf    = (const float*)d_in[12];
  const float* W_o1      = (const float*)d_in[13];
  const float* b_o1      = (const float*)d_in[14];
  const float* W_out     = (const float*)d_in[15];
  const float* W_id1     = (const float*)d_in[16];
  const float* b_id1     = (const float*)d_in[17];
  const float* W_id2     = (const float*)d_in[18];
  const float* W_ji      = (const float*)d_in[19];
  const float* b_ji      = (const float*)d_in[20];
  const float* W_kj      = (const float*)d_in[21];
  const float* b_kj      = (const float*)d_in[22];
  const float* W_irbf    = (const float*)d_in[23];
  const float* W_abf     = (const float*)d_in[24];
  const float* W_res     = (const float*)d_in[25];
  const float* b_res     = (const float*)d_in[26];
  const float* coef_mp   = (const float*)d_in[27];
  const float* coef_sg   = (const float*)d_in[28];

  const int na = in_sizes[0];          // N_ATOMS
  const int ne = in_sizes[3];          // N_EDGES
  const int nt = in_sizes[5];          // N_TRIP

  // workspace layout (fp32), 16B aligned chunks
  float* ws = (float*)d_ws;
  size_t off = 0;
  auto walloc = [&](size_t n) { float* p = ws + off; off += (n + 3) & ~(size_t)3; return p; };
  float* RBF  = walloc((size_t)ne * 6);     // [ne,6]
  float* EB0  = walloc((size_t)ne * EDIM);  // x (embedding accum -> edge state)
  float* EB1  = walloc((size_t)ne * EDIM);  // rbf_e, later x_ji
  float* EB2  = walloc((size_t)ne * EDIM);  // x_kj
  float* EB3  = walloc((size_t)ne * EDIM);  // aggr / h
  float* X0   = walloc((size_t)na * EDIM);  // atom embeddings
  float* PATM = walloc((size_t)na * EDIM);  // output-block segment sum
  float* HATM = walloc((size_t)na * EDIM);
  float* ROUT = walloc((size_t)na * EDIM);  // res_output (per atom)
  float* RSNG = walloc((size_t)na * EDIM);  // res_single (per atom)
  float* TMPA = walloc((size_t)na * EDIM);
  float* WPK  = walloc((size_t)EDIM * EDIM); // packed-W staging (64KB)

  const dim3 b256(256);
  const int g_edge32 = cdiv_i((long)ne * 32, 256);
  const int g_atom32 = cdiv_i((long)na * 32, 256);
  const int gg = cdiv_i(ne, 64);   // edge-level GEMM blocks
  const int ga = cdiv_i(na, 64);   // atom-level GEMM blocks
  const int gp = cdiv_i(64 * EDIM, 256);   // pack kernel blocks

  // helper: pack W then run GEMM with the packed buffer (stream-ordered)
  auto gemm = [&](const float* Asrc, const int* aidx, const float* W,
                  const float* bias, float* Cdst, const float* aux,
                  int nrows, int mode) {
    pack_w_kernel<<<gp, b256, 0, stream>>>(W, WPK);
    const int grid = cdiv_i(nrows, 64);
    gemm128_wmma_kernel<<<grid, b256, 0, stream>>>(Asrc, aidx, WPK, bias,
                                                   Cdst, aux, nrows, mode);
  };

  // --- basis functions ---
  edge_rbf_kernel<<<cdiv_i(ne, 256), b256, 0, stream>>>(R, idx_i, idx_j, RBF, ne);
  gather_rows_kernel<<<g_atom32, b256, 0, stream>>>(emb_atom, Z, X0, na);
  edge_radial_kernel<<<g_edge32, b256, 0, stream>>>(RBF, W_rbf_emb, nullptr, nullptr, EB1, ne, 0);

  // --- embedding block: x = silu([x0_i, x0_j, rbf_e] @ W_emb + b_emb) ---
  gemm(X0, idx_i, W_emb,               nullptr, EB0, nullptr, ne, 0);
  gemm(X0, idx_j, W_emb + 128 * EDIM,  nullptr, EB0, nullptr, ne, 1);
  gemm(EB1, nullptr, W_emb + 256 * EDIM, b_emb, EB0, nullptr, ne, 4);

  // --- output block 0 ---
  hipMemsetAsync(PATM, 0, (size_t)na * EDIM * 4, stream);
  edge_radial_kernel<<<g_edge32, b256, 0, stream>>>(RBF, W_orbf, EB0, idx_i, PATM, ne, 2);
  gemm(PATM, nullptr, W_o1, b_o1, HATM, nullptr, na, 2);
  gemm(HATM, nullptr, W_out, nullptr, ROUT, nullptr, na, 0);

  // --- identity block 0 ---
  gemm(X0, nullptr, W_id1, b_id1, TMPA, nullptr, na, 2);
  gemm(TMPA, nullptr, W_id2, nullptr, RSNG, nullptr, na, 0);

  for (int i = 0; i < 3; ++i) {
    const size_t mo = (size_t)i * EDIM * EDIM;
    const int ob = i + 1;
    const size_t mob = (size_t)ob * EDIM * EDIM;

    // interaction block i
    gemm(EB0, nullptr, W_ji + mo, b_ji + i * EDIM, EB1, nullptr, ne, 2);
    gemm(EB0, nullptr, W_kj + mo, b_kj + i * EDIM, EB2, nullptr, ne, 2);
    edge_radial_kernel<<<g_edge32, b256, 0, stream>>>(RBF, W_irbf + (size_t)i * 6 * EDIM, EB2, nullptr, EB2, ne, 1);
    hipMemsetAsync(EB3, 0, (size_t)ne * EDIM * 4, stream);
    triplet_kernel<<<cdiv_i((long)nt * 32, 256), b256, 0, stream>>>(
        EB2, W_abf + (size_t)i * 7 * EDIM, cosine, idx_kj, idx_ji, EB3, nt);
    ew_add_kernel<<<cdiv_i((long)ne * EDIM / 4, 256), b256, 0, stream>>>(EB3, EB1, (int)((long)ne * EDIM / 4));
    // x_next = h + silu(h @ W_res + b_res)
    gemm(EB3, nullptr, W_res + mo, b_res + i * EDIM, EB0, EB3, ne, 3);

    // output block i+1
    hipMemsetAsync(PATM, 0, (size_t)na * EDIM * 4, stream);
    edge_radial_kernel<<<g_edge32, b256, 0, stream>>>(RBF, W_orbf + (size_t)ob * 6 * EDIM, EB0, idx_i, PATM, ne, 2);
    gemm(PATM, nullptr, W_o1 + mob, b_o1 + ob * EDIM, HATM, nullptr, na, 2);
    gemm(HATM, nullptr, W_out + mob, nullptr, ROUT, nullptr, na, 1);

    // identity block i+1 (input = current res_single)
    gemm(RSNG, nullptr, W_id1 + mob, b_id1 + ob * EDIM, TMPA, nullptr, na, 2);
    gemm(TMPA, nullptr, W_id2 + mob, nullptr, RSNG, nullptr, na, 1);
  }

  // --- molecule reduction ---
  hipMemsetAsync(d_out, 0, (size_t)out_size * sizeof(float), stream);
  final_kernel<<<g_atom32, b256, 0, stream>>>(ROUT, RSNG, batch_seg, coef_mp, coef_sg,
                                              (float*)d_out, na);
}